// upc_sdg_21741124453037
// MI455X (gfx1250) — compile-verified
//
#include <hip/hip_runtime.h>
#include <hip/hip_bf16.h>
#include <cmath>

typedef __attribute__((ext_vector_type(2))) float v2f;
typedef __attribute__((ext_vector_type(8))) float v8f;

#define N_ITEMS   25000
#define EMB_D     64
#define BATCH     64
#define SEQ_L     50
#define NROWS     (BATCH * SEQ_L)     /* 3200 */
#define NEG_SLOPE 0.01f
#define NWAVES    8                   /* waves per workgroup in the big kernel */

/* d_out layout (floats, reference return order, flattened) */
#define OUT_USER  0
#define OUT_ITEM  (OUT_USER + BATCH * EMB_D)     /* 4096   */
#define OUT_ATT   (OUT_ITEM + NROWS * EMB_D)     /* 208896 */
#define OUT_IDX   (OUT_ATT + NROWS)              /* 212096 */
#define OUT_REPL  (OUT_IDX + NROWS)              /* 215296 */

/* d_ws layout (floats) */
#define WS_LATENT 0
#define WS_ATT    (WS_LATENT + NROWS * EMB_D)    /* 204800 */
#define WS_BANNED (WS_ATT + NROWS)               /* 208000; + 64*25000 floats = 7.2 MB total */

/* ---------------- small prep kernels (VALU; negligible cost) ---------------- */

__global__ void gather_user_kernel(const int* __restrict__ uidx,
                                   const float* __restrict__ uemb,
                                   float* __restrict__ out) {
  int i = blockIdx.x * blockDim.x + threadIdx.x;
  if (i < BATCH * EMB_D) {
    int b = i >> 6, d = i & 63;
    out[i] = uemb[(size_t)uidx[b] * EMB_D + d];
  }
}

__global__ void gather_item_kernel(const int* __restrict__ iidx,
                                   const float* __restrict__ iemb,
                                   float* __restrict__ out) {
  int i = blockIdx.x * blockDim.x + threadIdx.x;
  if (i < NROWS * EMB_D) {
    int r = i >> 6, d = i & 63;
    out[i] = iemb[(size_t)iidx[r] * EMB_D + d];
  }
}

__global__ void att_raw_kernel(const int* __restrict__ uidx,
                               const int* __restrict__ iidx,
                               const int* __restrict__ mask,
                               const float* __restrict__ uemb,
                               const float* __restrict__ iemb,
                               const float* __restrict__ Wsel,
                               const float* __restrict__ bsel,
                               float* __restrict__ ws_att) {
  int r = blockIdx.x * blockDim.x + threadIdx.x;
  if (r >= NROWS) return;
  int b = r / SEQ_L;
  const float* u  = uemb + (size_t)uidx[b] * EMB_D;
  const float* it = iemb + (size_t)iidx[r] * EMB_D;
  float acc = bsel[0];
  #pragma unroll 4
  for (int k = 0; k < EMB_D; ++k) acc += u[k] * Wsel[k];
  #pragma unroll 4
  for (int k = 0; k < EMB_D; ++k) acc += it[k] * Wsel[EMB_D + k];
  acc = (acc >= 0.f) ? acc : NEG_SLOPE * acc;               /* LeakyReLU */
  ws_att[r] = mask[r] ? acc : -INFINITY;
}

__global__ void att_softmax_kernel(const float* __restrict__ ws_att,
                                   float* __restrict__ out_att) {
  int b = threadIdx.x;
  if (b >= BATCH) return;
  float mm = -INFINITY;
  for (int l = 0; l < SEQ_L; ++l) mm = fmaxf(mm, ws_att[b * SEQ_L + l]);
  float ss = 0.f;
  for (int l = 0; l < SEQ_L; ++l) ss += __expf(ws_att[b * SEQ_L + l] - mm);
  float inv = 1.f / ss;
  for (int l = 0; l < SEQ_L; ++l)
    out_att[b * SEQ_L + l] = __expf(ws_att[b * SEQ_L + l] - mm) * inv;
}

__global__ __launch_bounds__(64) void latent_kernel(const int* __restrict__ uidx,
                                                    const int* __restrict__ iidx,
                                                    const float* __restrict__ priv,
                                                    const float* __restrict__ uemb,
                                                    const float* __restrict__ iemb,
                                                    const float* __restrict__ Wgen,
                                                    const float* __restrict__ bgen,
                                                    float* __restrict__ ws_latent) {
  int r = blockIdx.x, d = threadIdx.x;
  int b = r / SEQ_L;
  __shared__ float comb[2 * EMB_D];
  comb[d]         = uemb[(size_t)uidx[b] * EMB_D + d];
  comb[EMB_D + d] = iemb[(size_t)iidx[r] * EMB_D + d];
  __syncthreads();
  float acc = bgen[d] + priv[b] * Wgen[(size_t)(2 * EMB_D) * EMB_D + d];
  #pragma unroll 4
  for (int k = 0; k < 2 * EMB_D; ++k) acc += comb[k] * Wgen[(size_t)k * EMB_D + d];
  ws_latent[(size_t)r * EMB_D + d] = acc;
}

__global__ void banned_init_kernel(float* __restrict__ banned) {
  int i = blockIdx.x * blockDim.x + threadIdx.x;
  if (i < BATCH * N_ITEMS) banned[i] = 0.f;
}

__global__ void banned_scatter_kernel(const int* __restrict__ iidx,
                                      const int* __restrict__ mask,
                                      float* __restrict__ banned) {
  int r = blockIdx.x * blockDim.x + threadIdx.x;
  if (r < NROWS && mask[r])
    banned[(size_t)(r / SEQ_L) * N_ITEMS + iidx[r]] = -INFINITY;
}

/* ---------------- fused WMMA GEMM + gumbel + softmax/argmax ----------------
 *
 * One workgroup per 16-row tile (200 WGs, 8 waves each). Each wave owns the
 * A fragment (16x64 latent, registers) and strides over 16-column item tiles:
 * clause of 16 B-fragment loads + 8 NT gumbel loads + banned loads, then 16
 * back-to-back v_wmma_f32_16x16x4_f32, then a BRANCHLESS epilogue.
 *
 * Softmax without online rescaling: logits = sim + gumbel lie in ~[-3, 19]
 * (|sim| ~ 0.05, Gumbel(0,1) over 8e7 draws), so exp(x) in [5e-2, 7e8] is
 * safely inside fp32 range. Accumulate s = sum(exp(x)) directly, track
 * (max, argmax) with selects, finalize y* = exp(max)/s. One v_exp_f32 per
 * element, zero EXEC-mask divergence in the hot loop.
 */
__global__ __launch_bounds__(32 * NWAVES) void sim_softmax_kernel(
    const float* __restrict__ latent,    /* [NROWS, 64] (ws)        */
    const float* __restrict__ banned,    /* [BATCH, N_ITEMS] (ws)   */
    const float* __restrict__ item_emb,  /* [N_ITEMS, 64]           */
    const float* __restrict__ gumbel,    /* [NROWS, N_ITEMS]        */
    float* __restrict__ out_idx,         /* [NROWS]                 */
    float* __restrict__ out_repl)        /* [NROWS, 64]             */
{
  const int tile_r = blockIdx.x * 16;
  const int tid  = threadIdx.x;
  const int wave = tid >> 5;
  const int lane = tid & 31;
  const int half = lane >> 4;   /* 0: lanes 0-15, 1: lanes 16-31 */
  const int lm   = lane & 15;

  /* A fragment: per ISA A16x4 layout, lane holds M=lm, K = 4i + 2*half + {0,1}. */
  v2f a[16];
  const float* arow = latent + (size_t)(tile_r + lm) * EMB_D + 2 * half;
  #pragma unroll
  for (int i = 0; i < 16; ++i) a[i] = *(const v2f*)(arow + 4 * i);

  /* Rows owned by this lane: r = tile_r + v + 8*half, v = 0..7 (C-layout). */
  const int r0 = tile_r + 8 * half;
  const int b_lo = r0 / SEQ_L;
  const int b_hi = (r0 + 7) / SEQ_L;     /* at most 2 distinct batches per 8 rows */
  bool useLo[8];
  const float* grow[8];
  #pragma unroll
  for (int v = 0; v < 8; ++v) {
    useLo[v] = ((r0 + v) / SEQ_L) == b_lo;
    grow[v]  = gumbel + (size_t)(r0 + v) * N_ITEMS;
  }
  const float* ban_lo = banned + (size_t)b_lo * N_ITEMS;
  const float* ban_hi = banned + (size_t)b_hi * N_ITEMS;

  float m[8], s[8];
  int   am[8];
  #pragma unroll
  for (int v = 0; v < 8; ++v) { m[v] = -INFINITY; s[v] = 0.f; am[v] = 0; }

  const int NT = (N_ITEMS + 15) / 16;   /* 1563 column tiles */
  for (int t = wave; t < NT; t += NWAVES) {
    const int n   = t * 16 + lm;
    const int ncl = (n < N_ITEMS) ? n : (N_ITEMS - 1);  /* clamp: straight-line body */

    /* 1) clause of B-fragment loads into registers */
    const float* brow = item_emb + (size_t)ncl * EMB_D + 2 * half;
    v2f bb[16];
    #pragma unroll
    for (int i = 0; i < 16; ++i) bb[i] = *(const v2f*)(brow + 4 * i);

    /* 2) gumbel (streaming, non-temporal) + banned loads — overlap with WMMA */
    float g[8];
    #pragma unroll
    for (int v = 0; v < 8; ++v) g[v] = __builtin_nontemporal_load(grow[v] + ncl);
    const float bnLo = ban_lo[ncl];
    const float bnHi = (b_hi == b_lo) ? bnLo : ban_hi[ncl];

    /* 3) prefetch next tile's gumbel lines (clamped address, unconditional) */
    {
      const int np  = n + 16 * NWAVES;
      const int npc = (np < N_ITEMS) ? np : (N_ITEMS - 1);
      #pragma unroll
      for (int v = 0; v < 8; ++v) __builtin_prefetch(grow[v] + npc, 0, 1);
    }

    /* 4) 16 chained fp32 WMMAs: sim tile = A(16x64) x B(64x16) */
    v8f c = {};
    #pragma unroll
    for (int i = 0; i < 16; ++i)
      c = __builtin_amdgcn_wmma_f32_16x16x4_f32(false, a[i], false, bb[i],
                                                (short)0, c, false, false);

    /* 5) branchless epilogue: logits, running (max, argmax), sum of exp */
    const bool valid = (n < N_ITEMS);
    #pragma unroll
    for (int v = 0; v < 8; ++v) {
      const float bn = useLo[v] ? bnLo : bnHi;
      const float xr = c[v] + g[v] + bn;
      const float x  = valid ? xr : -INFINITY;   /* exp(-inf)=0, never wins max */
      const bool  gt = x > m[v];
      am[v] = gt ? n : am[v];
      m[v]  = gt ? x : m[v];
      s[v] += __expf(x);
    }
  }

  /* Merge across the 16 lanes that share each row (xor stays within a half). */
  #pragma unroll
  for (int v = 0; v < 8; ++v) {
    float mm = m[v], ss = s[v];
    int   aa = am[v];
    #pragma unroll
    for (int off = 1; off < 16; off <<= 1) {
      const float m2 = __shfl_xor(mm, off, 32);
      const float s2 = __shfl_xor(ss, off, 32);
      const int   a2 = __shfl_xor(aa, off, 32);
      const bool take = (m2 > mm) || (m2 == mm && a2 < aa);
      aa = take ? a2 : aa;
      mm = take ? m2 : mm;
      ss += s2;
    }
    m[v] = mm; s[v] = ss; am[v] = aa;
  }

  /* Merge across waves via LDS, then finalize. */
  __shared__ float lds_m[NWAVES][16], lds_s[NWAVES][16];
  __shared__ int   lds_a[NWAVES][16];
  __shared__ int   fin_arg[16];
  __shared__ float fin_scale[16];

  if (lm == 0) {
    #pragma unroll
    for (int v = 0; v < 8; ++v) {
      const int row = v + 8 * half;
      lds_m[wave][row] = m[v];
      lds_s[wave][row] = s[v];
      lds_a[wave][row] = am[v];
    }
  }
  __syncthreads();

  if (tid < 16) {
    float mm = lds_m[0][tid], ss = lds_s[0][tid];
    int   aa = lds_a[0][tid];
    #pragma unroll
    for (int w = 1; w < NWAVES; ++w) {
      const float m2 = lds_m[w][tid], s2 = lds_s[w][tid];
      const int   a2 = lds_a[w][tid];
      const bool take = (m2 > mm) || (m2 == mm && a2 < aa);
      aa = take ? a2 : aa;
      mm = take ? m2 : mm;
      ss += s2;
    }
    const float y     = __expf(mm) / ss;     /* y_soft at argmax */
    const float scale = (1.f + y) - y;       /* straight-through value, exact */
    out_idx[tile_r + tid] = (float)aa;
    fin_arg[tid]   = aa;
    fin_scale[tid] = scale;
  }
  __syncthreads();

  for (int i = tid; i < 16 * EMB_D; i += 32 * NWAVES) {
    const int rl = i >> 6, d = i & 63;
    out_repl[(size_t)(tile_r + rl) * EMB_D + d] =
        fin_scale[rl] * item_emb[(size_t)fin_arg[rl] * EMB_D + d];
  }
}

/* ---------------- launch ---------------- */

extern "C" void kernel_launch(void* const* d_in, const int* in_sizes, int n_in,
                              void* d_out, int out_size, void* d_ws, size_t ws_size,
                              hipStream_t stream) {
  const int*   user_indices = (const int*)d_in[0];
  const int*   item_indices = (const int*)d_in[1];
  const float* priv         = (const float*)d_in[2];
  const int*   mask         = (const int*)d_in[3];
  const float* uemb         = (const float*)d_in[4];
  const float* iemb         = (const float*)d_in[5];
  const float* Wsel         = (const float*)d_in[6];
  const float* bsel         = (const float*)d_in[7];
  const float* Wgen         = (const float*)d_in[8];
  const float* bgen         = (const float*)d_in[9];
  const float* gumbel       = (const float*)d_in[10];

  float* out = (float*)d_out;
  float* ws  = (float*)d_ws;
  float* ws_latent = ws + WS_LATENT;
  float* ws_att    = ws + WS_ATT;
  float* ws_banned = ws + WS_BANNED;

  gather_user_kernel<<<(BATCH * EMB_D + 255) / 256, 256, 0, stream>>>(
      user_indices, uemb, out + OUT_USER);
  gather_item_kernel<<<(NROWS * EMB_D + 255) / 256, 256, 0, stream>>>(
      item_indices, iemb, out + OUT_ITEM);
  att_raw_kernel<<<(NROWS + 255) / 256, 256, 0, stream>>>(
      user_indices, item_indices, mask, uemb, iemb, Wsel, bsel, ws_att);
  att_softmax_kernel<<<1, 64, 0, stream>>>(ws_att, out + OUT_ATT);
  latent_kernel<<<NROWS, 64, 0, stream>>>(
      user_indices, item_indices, priv, uemb, iemb, Wgen, bgen, ws_latent);
  banned_init_kernel<<<(BATCH * N_ITEMS + 255) / 256, 256, 0, stream>>>(ws_banned);
  banned_scatter_kernel<<<(NROWS + 255) / 256, 256, 0, stream>>>(
      item_indices, mask, ws_banned);
  sim_softmax_kernel<<<NROWS / 16, 32 * NWAVES, 0, stream>>>(
      ws_latent, ws_banned, iemb, gumbel, out + OUT_IDX, out + OUT_REPL);
}